// EmbCatDense_53309134078326
// MI455X (gfx1250) — compile-verified
//
#include <hip/hip_runtime.h>

// Problem constants (from reference)
#define T_TABLES 26
#define V_VOCAB  1000
#define D_DIM    128
#define B_BATCH  8192
#define L_BAG    8
#define OUT_W    ((T_TABLES + 1) * D_DIM)   // 3456 floats per output row

// ---- CDNA5 async global->LDS path detection --------------------------------
#if defined(__has_builtin)
#if __has_builtin(__builtin_amdgcn_global_load_async_to_lds_b128)
#define USE_ASYNC_LDS 1
#endif
#endif

// Builtin params: pointers to 16-byte int vectors in AS1 (global) / AS3 (LDS).
typedef int v4i __attribute__((vector_size(16)));
typedef float v4f __attribute__((vector_size(16)));
typedef __attribute__((address_space(1))) v4i g_v4i;
typedef __attribute__((address_space(3))) v4i l_v4i;

__device__ __forceinline__ void wait_asynccnt0() {
#if defined(__has_builtin) && __has_builtin(__builtin_amdgcn_s_wait_asynccnt)
  __builtin_amdgcn_s_wait_asynccnt(0);
#else
  asm volatile("s_wait_asynccnt 0" ::: "memory");
#endif
}

// One wave (32 lanes) per output segment.
// Segment s in [0, (T+1)*B): b = s % B, t1 = s / B.
//   t1 == 0      -> copy to_cat row b into out[b, 0:128]
//   t1 in [1,26] -> pool table t = t1-1, bag b, into out[b, t1*128 : +128]
// Each lane owns one float4: a wave covers the 128-float segment with one
// b128 op per row -> 512B coalesced transactions.
//
// Cache policy: embedding-table reads stay RT (13MB of tables vs 192MB L2 ->
// the 872MB of gathered rows are L2 hits). The 113MB write-once output and
// the read-once to_cat stream are marked non-temporal so they don't evict
// the tables from L2.
__global__ __launch_bounds__(256) void EmbCatDense_pool_kernel(
    const int*   __restrict__ indices,   // [T, B, L] int32
    const float* __restrict__ to_cat,    // [B, D]
    const float* __restrict__ emb,       // [T, V, D]
    float*       __restrict__ out)       // [B, (T+1)*D]
{
  const int lane = threadIdx.x & 31;
  const int wave = threadIdx.x >> 5;
  const int seg  = blockIdx.x * 8 + wave;          // 0 .. 27*8192-1 (exact)
  const int b    = seg & (B_BATCH - 1);            // B = 2^13
  const int t1   = seg >> 13;

  v4f* outp = (v4f*)(out + (size_t)b * OUT_W + (size_t)t1 * D_DIM + lane * 4);

  if (t1 == 0) {
    // Dense passthrough: pure NT copy, 512B/wave.
    const v4f v = __builtin_nontemporal_load(
        (const v4f*)(to_cat + (size_t)b * D_DIM + lane * 4));
    __builtin_nontemporal_store(v, outp);
    return;
  }

  const int t = t1 - 1;
  // 8 bag indices, wave-uniform address -> single L0 request, lanes broadcast.
  const int* ip = indices + ((size_t)t * B_BATCH + b) * L_BAG;
  const int4 i0 = *(const int4*)(ip);
  const int4 i1 = *(const int4*)(ip + 4);
  const int rows[L_BAG] = {i0.x, i0.y, i0.z, i0.w, i1.x, i1.y, i1.z, i1.w};

  const float* tab = emb + (size_t)t * (V_VOCAB * D_DIM);

  v4f acc = {0.f, 0.f, 0.f, 0.f};

#if USE_ASYNC_LDS
  // 8 waves * 8 rows * 128 floats = 32 KB LDS per block.
  // One GLOBAL_LOAD_ASYNC_TO_LDS_B128 fetches one full 512B embedding row
  // (32 lanes x 16B) straight into LDS, tracked on ASYNCcnt -- no VGPR
  // landing zone, so the 8-row gather is fully in flight at once.
  __shared__ float lds[8 * L_BAG * D_DIM];
  float* myLds = &lds[wave * (L_BAG * D_DIM)];

#pragma unroll
  for (int r = 0; r < L_BAG; ++r) {
    const float* gsrc = tab + (size_t)rows[r] * D_DIM + lane * 4;
    float*       ldst = myLds + r * D_DIM + lane * 4;
    __builtin_amdgcn_global_load_async_to_lds_b128(
        (g_v4i*)gsrc, (l_v4i*)ldst, /*offset=*/0, /*cpol=*/0);
  }

  wait_asynccnt0();   // s_wait_asynccnt 0: all 8 rows resident in LDS

#pragma unroll
  for (int r = 0; r < L_BAG; ++r) {
    const v4f v = *(const v4f*)(myLds + r * D_DIM + lane * 4);
    acc += v;
  }
#else
  // Fallback: direct b128 gathers (rows are L2-resident).
#pragma unroll
  for (int r = 0; r < L_BAG; ++r) {
    const v4f v = *(const v4f*)(tab + (size_t)rows[r] * D_DIM + lane * 4);
    acc += v;
  }
#endif

  __builtin_nontemporal_store(acc, outp);  // b128 store, TH=NT (write-once)
}

extern "C" void kernel_launch(void* const* d_in, const int* in_sizes, int n_in,
                              void* d_out, int out_size, void* d_ws, size_t ws_size,
                              hipStream_t stream) {
  const int*   indices = (const int*)  d_in[0];   // [T,B,L]
  // d_in[1] = offsets: encodes fixed bag length L, unused.
  const float* to_cat  = (const float*)d_in[2];   // [B,D]
  const float* emb     = (const float*)d_in[3];   // [T,V,D]
  float*       out     = (float*)d_out;           // [B,(T+1)*D]

  // (T+1)*B segments, 8 waves (256 threads) per block -> exact grid.
  const int nseg   = (T_TABLES + 1) * B_BATCH;    // 27*8192 = 221184
  const int blocks = nseg / 8;                    // 27648
  EmbCatDense_pool_kernel<<<blocks, 256, 0, stream>>>(indices, to_cat, emb, out);
}